// HashGridEncode_44555990729142
// MI455X (gfx1250) — compile-verified
//
#include <hip/hip_runtime.h>
#include <hip/hip_bf16.h>
#include <stdint.h>

typedef __attribute__((ext_vector_type(2))) float v2f;
typedef __attribute__((ext_vector_type(4))) float v4f;
typedef __attribute__((ext_vector_type(2))) int   v2i;

// Packed per-level constants: {res+1, base offset}. dense == (level < 7),
// all hashed levels have size 2^19 (mask 0x7FFFF), resf = (float)(res1-1).
__device__ __constant__ v2i c_lvl[16] = {
    {17, 0},        {21, 4913},     {26, 14174},    {33, 31750},
    {41, 67687},    {51, 136608},   {65, 269259},   {81, 543884},
    {102, 1068172}, {129, 1592460}, {162, 2116748}, {204, 2641036},
    {257, 3165324}, {323, 3689612}, {407, 4213900}, {513, 4738188}
};

#define HASH_MASK 0x7FFFFu
#define PRIME1 2654435761u
#define PRIME2 805459861u

__global__ __launch_bounds__(256)
void HashGridEncode_44555990729142_kernel(const float* __restrict__ xyz,
                                          const float* __restrict__ emb,
                                          const float* __restrict__ mn,
                                          const float* __restrict__ mx,
                                          float* __restrict__ out,
                                          int Btot)
{
    __shared__ float sxyz[48];      // 16 points x 3 raw coords (async staged)
    __shared__ v4f   snorm[16];     // per-point {nx, ny, nz, validMask}

    const int t  = threadIdx.x;     // 0..255
    const int b0 = blockIdx.x * 16; // first point of this block

    // ---- CDNA5 async global->LDS staging of this block's xyz (48 dwords) ----
    if (t < 48 && (b0 * 3 + t) < Btot * 3) {
        unsigned lds_off = (unsigned)(uintptr_t)(&sxyz[t]);   // low 32 bits = LDS byte offset
        unsigned goff    = (unsigned)((b0 * 3 + t) * 4);
        unsigned long long sbase = (unsigned long long)(uintptr_t)xyz;
        asm volatile("global_load_async_to_lds_b32 %0, %1, %2"
                     :: "v"(lds_off), "v"(goff), "s"(sbase)
                     : "memory");
    }
    asm volatile("s_wait_asynccnt 0" ::: "memory");
    __syncthreads();

    // ---- normalize each point exactly once (16 workers), publish via LDS ----
    if (t < 16) {
        const float px = sxyz[t*3+0], py = sxyz[t*3+1], pz = sxyz[t*3+2];
        const float mn0 = mn[0], mn1 = mn[1], mn2 = mn[2];
        float nx = (px - mn0) / (mx[0] - mn0);
        float ny = (py - mn1) / (mx[1] - mn1);
        float nz = (pz - mn2) / (mx[2] - mn2);
        const bool valid = (nx >= 0.f) & (nx <= 1.f) & (ny >= 0.f) & (ny <= 1.f)
                         & (nz >= 0.f) & (nz <= 1.f);
        v4f s;
        s.x = fminf(fmaxf(nx, 0.f), 1.f);
        s.y = fminf(fmaxf(ny, 0.f), 1.f);
        s.z = fminf(fmaxf(nz, 0.f), 1.f);
        s.w = valid ? 1.f : 0.f;
        snorm[t] = s;
    }
    __syncthreads();

    const int p = t >> 4;           // point within block
    const int l = t & 15;           // level (fastest -> coalesced output)
    const int b = b0 + p;
    if (b >= Btot) return;

    const v4f  nrm = snorm[p];                  // one ds_load_b128
    const v2i  lvl = c_lvl[l];                  // one b64 constant fetch
    const int  r1       = lvl.x;                // res + 1
    const int  base_off = lvl.y;
    const bool dense    = (l < 7);
    const float res = (float)(r1 - 1);          // one v_cvt instead of a table

    const float sx = nrm.x * res, sy = nrm.y * res, sz = nrm.z * res;
    const float bfx = fminf(floorf(sx), res - 1.f);
    const float bfy = fminf(floorf(sy), res - 1.f);
    const float bfz = fminf(floorf(sz), res - 1.f);
    const float fx = sx - bfx, fy = sy - bfy, fz = sz - bfz;
    const int bx = (int)bfx, by = (int)bfy, bz = (int)bfz;

    // trilinear weights, corner g bits = (gx,gy,gz) matching meshgrid 'ij'
    const float w0x = 1.f - fx, w0y = 1.f - fy, w0z = 1.f - fz;
    const float w00 = w0x * w0y, w01 = w0x * fy, w10 = fx * w0y, w11 = fx * fy;
    float wgt[8];
    wgt[0] = w00 * w0z; wgt[1] = w00 * fz;
    wgt[2] = w01 * w0z; wgt[3] = w01 * fz;
    wgt[4] = w10 * w0z; wgt[5] = w10 * fz;
    wgt[6] = w11 * w0z; wgt[7] = w11 * fz;

    // dense index deltas (levels 0..6) — zero-mod by construction
    const int d000 = (bx * r1 + by) * r1 + bz;
    const int dX = r1 * r1, dY = r1;

    // hash partials (levels 7..15), prime0 == 1 so x contributes directly
    const unsigned hx0 = (unsigned)bx,          hx1 = (unsigned)(bx + 1);
    const unsigned hy0 = (unsigned)by * PRIME1, hy1 = (unsigned)(by + 1) * PRIME1;
    const unsigned hz0 = (unsigned)bz * PRIME2, hz1 = (unsigned)(bz + 1) * PRIME2;

    int idx[8];
#pragma unroll
    for (int g = 0; g < 8; ++g) {
        const int gx = (g >> 2) & 1, gy = (g >> 1) & 1, gz = g & 1;
        const int      di = d000 + (gx ? dX : 0) + (gy ? dY : 0) + gz;
        const unsigned hi = (gx ? hx1 : hx0) ^ (gy ? hy1 : hy0) ^ (gz ? hz1 : hz0);
        idx[g] = base_off + (dense ? di : (int)(hi & HASH_MASK));
    }

    // issue all 8 float2 gathers (table is L2-resident: 42 MB < 192 MB), then reduce
    v2f f[8];
#pragma unroll
    for (int g = 0; g < 8; ++g)
        f[g] = *(const v2f*)(emb + 2 * (size_t)idx[g]);

    float a0 = 0.f, a1 = 0.f;
#pragma unroll
    for (int g = 0; g < 8; ++g) {
        a0 = fmaf(wgt[g], f[g].x, a0);
        a1 = fmaf(wgt[g], f[g].y, a1);
    }
    v2f o; o.x = a0 * nrm.w; o.y = a1 * nrm.w;

    // streaming (non-temporal) store: don't evict the embedding table from L2
    __builtin_nontemporal_store(o, (v2f*)out + (size_t)b * 16 + l);
}

extern "C" void kernel_launch(void* const* d_in, const int* in_sizes, int n_in,
                              void* d_out, int out_size, void* d_ws, size_t ws_size,
                              hipStream_t stream) {
    const float* xyz = (const float*)d_in[0];
    const float* emb = (const float*)d_in[1];
    const float* mn  = (const float*)d_in[2];
    const float* mx  = (const float*)d_in[3];
    float* out = (float*)d_out;

    const int Btot   = in_sizes[0] / 3;          // points
    const int blocks = (Btot + 15) / 16;         // 16 points per 256-thread block
    hipLaunchKernelGGL(HashGridEncode_44555990729142_kernel,
                       dim3(blocks), dim3(256), 0, stream,
                       xyz, emb, mn, mx, out, Btot);
}